// ModDrop_77077483094420
// MI455X (gfx1250) — compile-verified
//
#include <hip/hip_runtime.h>

// Problem geometry (fixed by the reference)
#define B_N    64
#define C_N    4
#define HW     262144          // 512*512 floats per (b,c) slab
#define HW4    65536           // float4 per slab
#define SPLIT  16              // partial-sum splits per slab
#define CHUNK4 4096            // float4 per (bc,split) chunk = HW4/SPLIT
#define TPB    256             // 8 waves (wave32)

typedef __attribute__((ext_vector_type(2))) float v2f;
typedef __attribute__((ext_vector_type(4))) float f32x4;
typedef __attribute__((ext_vector_type(8))) float v8f;

// group 0 = channels {0,1,2}, group 1 = {3}
__device__ __forceinline__ int channel_dropped(int drop, int choice, int c) {
    return drop && ((choice == 0) ? (c < 3) : (c == 3));
}

// Full 32-lane sum via V_WMMA_F32_16X16X4_F32 ones-trick.
// A[m][*] = {L(m), 0, L(m+16), 0}; B = ones(4x16) ->
// D[m][n] = L(m)+L(m+16). Sum the 8 D VGPRs (rows 0-7 land in lanes 0-15,
// rows 8-15 in lanes 16-31), then one cross-half add completes the reduce.
__device__ __forceinline__ float wave_sum_wmma(float v) {
    v2f a;  a.x = v;    a.y = 0.0f;
    v2f bo; bo.x = 1.0f; bo.y = 1.0f;
    v8f c = {};
    v8f d = __builtin_amdgcn_wmma_f32_16x16x4_f32(
        /*neg_a=*/false, a, /*neg_b=*/false, bo,
        /*c_mod=*/(short)0, c, /*reuse_a=*/false, /*reuse_b=*/false);
    float s = ((d[0] + d[1]) + (d[2] + d[3])) + ((d[4] + d[5]) + (d[6] + d[7]));
    s += __shfl_xor(s, 16, 32);   // combine lanes 0-15 half with 16-31 half
    return s;
}

// Pass 1: deterministic per-(b,c,split) partial sums. Dropped channels are
// never read (their sum is exactly 0 by construction).
__global__ void __launch_bounds__(TPB)
moddrop_partial_sums(const float* __restrict__ x,
                     const int* __restrict__ drop,
                     const int* __restrict__ choice,
                     float* __restrict__ partial) {
    const int bc = blockIdx.x >> 4;      // (b*4 + c)
    const int sp = blockIdx.x & (SPLIT - 1);
    const int b  = bc >> 2, c = bc & 3;

    if (channel_dropped(drop[b], choice[b], c)) {   // uniform branch, EXEC stays full
        if (threadIdx.x == 0) partial[blockIdx.x] = 0.0f;
        return;
    }

    const f32x4* x4 = (const f32x4*)x + (size_t)bc * HW4 + (size_t)sp * CHUNK4;
    float acc = 0.0f;
#pragma unroll
    for (int i = 0; i < CHUNK4 / TPB; ++i) {        // 16 x global_load_b128
        f32x4 v = x4[i * TPB + threadIdx.x];
        acc += (v.x + v.y) + (v.z + v.w);
    }

    const float ws = wave_sum_wmma(acc);            // EXEC all-1s here

    __shared__ float lsum[TPB / 32];
    const int lane = threadIdx.x & 31, wave = threadIdx.x >> 5;
    if (lane == 0) lsum[wave] = ws;
    __syncthreads();
    if (threadIdx.x == 0) {
        float t = 0.0f;
#pragma unroll
        for (int w = 0; w < TPB / 32; ++w) t += lsum[w];
        partial[blockIdx.x] = t;
    }
}

// Pass 1b: ch_sums -> gain[b] = #channels with nonzero spatial sum.
__global__ void __launch_bounds__(TPB)
moddrop_gain(const float* __restrict__ partial, float* __restrict__ gain) {
    __shared__ float nz[B_N * C_N];
    const int bc = threadIdx.x;                     // 0..255 == one (b,c) each
    float t = 0.0f;
#pragma unroll
    for (int s = 0; s < SPLIT; ++s) t += partial[bc * SPLIT + s];
    nz[bc] = (t != 0.0f) ? 1.0f : 0.0f;
    __syncthreads();
    if (threadIdx.x < B_N) {
        const int b = threadIdx.x;
        gain[b] = ((nz[4 * b] + nz[4 * b + 1]) + (nz[4 * b + 2] + nz[4 * b + 3]));
    }
}

// Pass 2: out = x * mask / gain. Dropped channels are written as zeros with
// no read. Last-use reads + write-once stores are non-temporal so the 256 MB
// of output traffic does not evict x from the 192 MB L2 between passes.
__global__ void __launch_bounds__(TPB)
moddrop_scale(const float* __restrict__ x,
              const int* __restrict__ drop,
              const int* __restrict__ choice,
              const float* __restrict__ gain,
              float* __restrict__ out) {
    const int bc  = blockIdx.x >> 8;                // 256 blocks per (b,c) slab
    const int blk = blockIdx.x & 255;
    const int b = bc >> 2, c = bc & 3;
    const size_t idx4 = (size_t)bc * HW4 + (size_t)blk * TPB + threadIdx.x;
    f32x4* o4 = (f32x4*)out;

    if (channel_dropped(drop[b], choice[b], c)) {
        f32x4 z = {0.0f, 0.0f, 0.0f, 0.0f};
        __builtin_nontemporal_store(z, &o4[idx4]);
        return;
    }
    const float inv = 1.0f / gain[b];               // gain in {1,3,4}
    f32x4 v = __builtin_nontemporal_load((const f32x4*)x + idx4);
    v.x *= inv; v.y *= inv; v.z *= inv; v.w *= inv;
    __builtin_nontemporal_store(v, &o4[idx4]);
}

extern "C" void kernel_launch(void* const* d_in, const int* in_sizes, int n_in,
                              void* d_out, int out_size, void* d_ws, size_t ws_size,
                              hipStream_t stream) {
    const float* x      = (const float*)d_in[0];   // [64,4,512,512] f32
    const int*   drop   = (const int*)d_in[1];     // [64] i32
    const int*   choice = (const int*)d_in[2];     // [64] i32
    float*       out    = (float*)d_out;

    float* partial = (float*)d_ws;                    // 4096 floats
    float* gain    = partial + B_N * C_N * SPLIT;     // 64 floats

    moddrop_partial_sums<<<B_N * C_N * SPLIT, TPB, 0, stream>>>(x, drop, choice, partial);
    moddrop_gain<<<1, TPB, 0, stream>>>(partial, gain);
    moddrop_scale<<<B_N * C_N * HW4 / TPB, TPB, 0, stream>>>(x, drop, choice, gain, out);
}